// Reverse_quantize_49134425866735
// MI455X (gfx1250) — compile-verified
//
#include <hip/hip_runtime.h>
#include <hip/hip_bf16.h>
#include <stdint.h>

// Reverse-quantize (VQ codebook lookup):
//   out[row, :] = embedding_weight[indices[row], :]   row in [0, 32768), E_DIM = 512 f32
//
// Pure gather, HBM-bandwidth bound (~16 MB table held in 192 MB L2, 64 MB out).
// Implemented with the CDNA5 async DMA path: global -> LDS -> global using
// GLOBAL_LOAD_ASYNC_TO_LDS_B128 / GLOBAL_STORE_ASYNC_FROM_LDS_B128 (ASYNCcnt),
// so row data never passes through VGPRs.

#define E_DIM            512
#define ROW_BYTES        (E_DIM * 4)        // 2048 bytes per row
#define WAVES_PER_BLOCK  8
#define BLOCK_THREADS    (WAVES_PER_BLOCK * 32)

__global__ __launch_bounds__(BLOCK_THREADS)
void vq_gather_async_kernel(const int* __restrict__ indices,
                            const float* __restrict__ emb,
                            float* __restrict__ out,
                            int n_rows)
{
    // One 2 KB staging buffer per wave (16 KB per block).
    __shared__ float stage[WAVES_PER_BLOCK * E_DIM];

    const int wave = threadIdx.x >> 5;     // gfx1250 is wave32
    const int lane = threadIdx.x & 31;
    const int row  = blockIdx.x * WAVES_PER_BLOCK + wave;
    if (row >= n_rows) return;

    // Wave-uniform index -> scalar load by the compiler.
    const int idx = indices[row];

    // LDS byte offset of this lane's 16-byte slot. Truncating the flat
    // (generic) address to 32 bits yields the LDS offset: the shared-aperture
    // bits live in addr[63:32] (ISA 10.2).
    const unsigned lds_lane =
        (unsigned)(uintptr_t)(&stage[wave * E_DIM]) + (unsigned)(lane * 16);

    // ---- Async load: codebook row -> LDS (4 x B128 per wave = 2048 B) ----
    // GVS addressing: mem = SGPR64(base) + VGPR32(offset) + IOFFSET.
    // IOFFSET is applied to BOTH the global and the LDS address, so a single
    // lds/global offset pair covers the whole row with immediate offsets.
    const unsigned gld = (unsigned)idx * (unsigned)ROW_BYTES + (unsigned)(lane * 16);
    asm volatile(
        "global_load_async_to_lds_b128 %0, %1, %2 offset:0\n\t"
        "global_load_async_to_lds_b128 %0, %1, %2 offset:512\n\t"
        "global_load_async_to_lds_b128 %0, %1, %2 offset:1024\n\t"
        "global_load_async_to_lds_b128 %0, %1, %2 offset:1536"
        :
        : "v"(lds_lane), "v"(gld), "s"(emb)
        : "memory");

    // Async loads and stores are NOT ordered at the LDS (ISA 4.1): wait for
    // the loads to land in LDS before the DMA store reads it.
    asm volatile("s_wait_asynccnt 0x0" ::: "memory");

    // ---- Async store: LDS -> out row (fire and forget) ----
    const unsigned gst = (unsigned)row * (unsigned)ROW_BYTES + (unsigned)(lane * 16);
    asm volatile(
        "global_store_async_from_lds_b128 %0, %1, %2 offset:0\n\t"
        "global_store_async_from_lds_b128 %0, %1, %2 offset:512\n\t"
        "global_store_async_from_lds_b128 %0, %1, %2 offset:1024\n\t"
        "global_store_async_from_lds_b128 %0, %1, %2 offset:1536"
        :
        : "v"(gst), "v"(lds_lane), "s"(out)
        : "memory");
    // S_ENDPGM performs an implicit wait-idle, draining the async stores
    // before the wave (and its LDS allocation) retires.
}

extern "C" void kernel_launch(void* const* d_in, const int* in_sizes, int n_in,
                              void* d_out, int out_size, void* d_ws, size_t ws_size,
                              hipStream_t stream)
{
    const int*   indices = (const int*)d_in[0];    // [N] int32
    const float* emb     = (const float*)d_in[1];  // [N_E, E_DIM] f32
    float*       out     = (float*)d_out;          // [N, E_DIM] f32

    const int n_rows = in_sizes[0];                // 32768
    const int blocks = (n_rows + WAVES_PER_BLOCK - 1) / WAVES_PER_BLOCK;

    vq_gather_async_kernel<<<blocks, BLOCK_THREADS, 0, stream>>>(
        indices, emb, out, n_rows);
}